// ContinuousSSMLayer_19155554140616
// MI455X (gfx1250) — compile-verified
//
#include <hip/hip_runtime.h>

// ---------------------------------------------------------------------------
// S5/S4D SSM layer for MI455X (gfx1250).
//   L=32768, H=256, P=128 (complex, conj-sym).  fp32 throughout.
// Memory-bound: ~160 MB HBM traffic -> ~7us @ 23.3 TB/s.  Compute ~8.6 GFLOP
// fp32 on V_WMMA_F32_16X16X4_F32; the scan recurrence runs on (dual-issue)
// VALU in the WMMA accumulator layout.
// LC=128 -> 256 workgroups (one 8-wave WG per WGP on a ~256-WGP part).
// Inter-chunk combine is a Kogge-Stone scan (8 steps) instead of a 256-long
// serial chain.
// ---------------------------------------------------------------------------

typedef __attribute__((ext_vector_type(2))) float v2f;
typedef __attribute__((ext_vector_type(8))) float v8f;

constexpr int L  = 32768;
constexpr int H  = 256;
constexpr int P  = 128;
constexpr int LC = 128;            // chunk length (one workgroup per chunk)
constexpr int NCHUNK = L / LC;     // 256

// workspace layout (in floats)
constexpr size_t OFF_BBT_RE = 0;                       // [H*P]  B_bar^T real, idx h*P+p
constexpr size_t OFF_BBT_IM = OFF_BBT_RE + H * P;      // [H*P]
constexpr size_t OFF_CRET   = OFF_BBT_IM + H * P;      // [P*H]  2*C_re^T, idx p*H+h
constexpr size_t OFF_CIMTN  = OFF_CRET + P * H;        // [P*H] -2*C_im^T
constexpr size_t OFF_ABAR   = OFF_CIMTN + P * H;       // [P] float2
constexpr size_t OFF_POWA   = OFF_ABAR + 2 * P;        // [LC*P] float2: powA[j*P+p]=Abar^(j+1)
constexpr size_t OFF_CEND   = OFF_POWA + 2 * LC * P;   // [NCHUNK*P] float2 local chunk end state
constexpr size_t OFF_CARRY  = OFF_CEND + 2 * NCHUNK * P;  // [NCHUNK*P] float2 carry into chunk
constexpr size_t OFF_XLOC   = OFF_CARRY + 2 * NCHUNK * P; // [L*P] float2 local-scan x

// D = A(16x4) * B(4x16) + C(16x16), all fp32, wave32.
__device__ __forceinline__ v8f wmma_f32(v2f a, v2f b, v8f c) {
  return __builtin_amdgcn_wmma_f32_16x16x4_f32(false, a, false, b, (short)0, c,
                                               false, false);
}

// ---------------------------------------------------------------------------
// K0: discretization + operand layout prep.
// ---------------------------------------------------------------------------
__global__ void k0_setup(const float* __restrict__ Lre,
                         const float* __restrict__ Lim,
                         const float* __restrict__ B,     // (P,H,2)
                         const float* __restrict__ C,     // (H,P,2)
                         const float* __restrict__ lstep, // (P,)
                         float* __restrict__ ws) {
  float* BbT_re = ws + OFF_BBT_RE;
  float* BbT_im = ws + OFF_BBT_IM;
  float* CreT   = ws + OFF_CRET;
  float* CimTn  = ws + OFF_CIMTN;
  float2* Abar  = (float2*)(ws + OFF_ABAR);
  float2* powA  = (float2*)(ws + OFF_POWA);

  const int tid = blockIdx.x * blockDim.x + threadIdx.x;
  const int nth = gridDim.x * blockDim.x;

  for (int i = tid; i < H * P; i += nth) {
    const int h = i / P, p = i % P;
    const float lr = Lre[p], li = Lim[p];
    const float dt = __expf(lstep[p]);
    const float zr = lr * dt, zi = li * dt;
    const float ea = __expf(zr);
    const float abr = ea * __cosf(zi), abi = ea * __sinf(zi);
    // (Abar - 1) / lambda   (complex divide)
    const float den = lr * lr + li * li;
    const float nr = abr - 1.0f, ni = abi;
    const float cr = (nr * lr + ni * li) / den;
    const float ci = (ni * lr - nr * li) / den;
    const float br = B[(p * H + h) * 2 + 0];
    const float bi = B[(p * H + h) * 2 + 1];
    BbT_re[h * P + p] = cr * br - ci * bi;
    BbT_im[h * P + p] = cr * bi + ci * br;
    if (h == 0) Abar[p] = make_float2(abr, abi);
    // transpose C with the conj-sym scaling folded in
    CreT[p * H + h]  =  2.0f * C[(h * P + p) * 2 + 0];
    CimTn[p * H + h] = -2.0f * C[(h * P + p) * 2 + 1];
  }

  // power table Abar^(j+1), j=0..LC-1 (block 0, one thread per p)
  if (blockIdx.x == 0 && threadIdx.x < P) {
    const int p = threadIdx.x;
    const float lr = Lre[p], li = Lim[p];
    const float dt = __expf(lstep[p]);
    const float ea = __expf(lr * dt);
    const float abr = ea * __cosf(li * dt), abi = ea * __sinf(li * dt);
    float xr = abr, xi = abi;
    for (int j = 0; j < LC; ++j) {
      powA[j * P + p] = make_float2(xr, xi);
      const float nxr = xr * abr - xi * abi;
      const float nxi = xr * abi + xi * abr;
      xr = nxr; xi = nxi;
    }
  }
}

// ---------------------------------------------------------------------------
// K1: per-chunk  Bu = u @ B_bar^T  (WMMA fp32)  +  in-chunk scan in the WMMA
//     accumulator layout.  One workgroup (8 waves) per 128-row chunk; wave w
//     owns state channels p in [16w, 16w+16).
// ---------------------------------------------------------------------------
__global__ __launch_bounds__(256) void k1_chunk(const float* __restrict__ u,
                                                float* __restrict__ ws) {
  const float* BbT_re = ws + OFF_BBT_RE;
  const float* BbT_im = ws + OFF_BBT_IM;
  const float2* Abar  = (const float2*)(ws + OFF_ABAR);
  float2* chunk_end   = (float2*)(ws + OFF_CEND);
  float2* xloc        = (float2*)(ws + OFF_XLOC);

  __shared__ float lds_u[16 * (H + 2)];  // padded stride -> bank spread

  const int chunk = blockIdx.x;
  const int tid   = threadIdx.x;
  const int wave  = tid >> 5;
  const int lane  = tid & 31;
  const int p     = wave * 16 + (lane & 15);
  const int khalf = (lane >> 4) * 2;     // A/B fragment K sub-offset (0 or 2)

  const float2 A = Abar[p];
  float cr = 0.0f, ci = 0.0f;            // in-chunk running carry (local scan)

  for (int mt = 0; mt < LC / 16; ++mt) {
    const int l0 = chunk * LC + mt * 16;

    __syncthreads();
    // stage 16 x H u-tile into LDS (float2 granularity)
    for (int i = tid; i < 16 * (H / 2); i += 256) {
      const int r  = i / (H / 2);
      const int c2 = (i % (H / 2)) * 2;
      const float2 v = *(const float2*)(u + (size_t)(l0 + r) * H + c2);
      lds_u[r * (H + 2) + c2]     = v.x;
      lds_u[r * (H + 2) + c2 + 1] = v.y;
    }
    if (mt + 1 < LC / 16) {  // pull next tile toward the caches
      __builtin_prefetch(u + (size_t)(l0 + 16) * H + (tid & 63) * 16, 0, 3);
    }
    __syncthreads();

    // Bu tile: acc(16x16) += u(16xH) @ B_bar^T(H x p-slice), re & im
    v8f acc_re = {0.f, 0.f, 0.f, 0.f, 0.f, 0.f, 0.f, 0.f};
    v8f acc_im = {0.f, 0.f, 0.f, 0.f, 0.f, 0.f, 0.f, 0.f};
    const float* arow = &lds_u[(lane & 15) * (H + 2)];
#pragma unroll 4
    for (int k0 = 0; k0 < H; k0 += 4) {
      v2f a;                              // A frag: lane holds (M=lane&15, K=k0+khalf .. +1)
      a.x = arow[k0 + khalf];
      a.y = arow[k0 + khalf + 1];
      const int rk = (k0 + khalf) * P + p;  // B frag: rows k, k+1 of B_bar^T
      v2f br_, bi_;
      br_.x = BbT_re[rk];  br_.y = BbT_re[rk + P];
      bi_.x = BbT_im[rk];  bi_.y = BbT_im[rk + P];
      acc_re = wmma_f32(a, br_, acc_re);
      acc_im = wmma_f32(a, bi_, acc_im);
    }

    // In-register scan across the tile's 16 rows.  Acc layout: vgpr r holds
    // row (r) for lanes 0-15 and row (r+8) for lanes 16-31, col = p.
    float xr = cr, xi = ci;
    if (lane < 16) {
#pragma unroll
      for (int r = 0; r < 8; ++r) {
        const float nr = A.x * xr - A.y * xi + acc_re[r];
        const float ni = A.x * xi + A.y * xr + acc_im[r];
        xr = nr; xi = ni;
        acc_re[r] = xr; acc_im[r] = xi;
      }
    }
    const float er = __shfl(xr, lane & 15, 32);   // row-7 state -> upper half
    const float ei = __shfl(xi, lane & 15, 32);
    if (lane >= 16) {
      xr = er; xi = ei;
#pragma unroll
      for (int r = 0; r < 8; ++r) {
        const float nr = A.x * xr - A.y * xi + acc_re[r];
        const float ni = A.x * xi + A.y * xr + acc_im[r];
        xr = nr; xi = ni;
        acc_re[r] = xr; acc_im[r] = xi;
      }
    }
    cr = __shfl(xr, (lane & 15) + 16, 32);        // row-15 state -> everyone
    ci = __shfl(xi, (lane & 15) + 16, 32);

    // spill local-scan x tile (float2 interleaved re/im)
    const int rowoff = (lane >> 4) * 8;
#pragma unroll
    for (int r = 0; r < 8; ++r) {
      const int l = l0 + rowoff + r;
      xloc[(size_t)l * P + p] = make_float2(acc_re[r], acc_im[r]);
    }
  }

  if (lane < 16) chunk_end[chunk * P + p] = make_float2(cr, ci);
}

// ---------------------------------------------------------------------------
// K2: inter-chunk Kogge-Stone scan.  One block per state channel p; thread t
// holds chunk t's element (A = Abar^LC, b = local chunk-end state).
// lin_op composition (matches reference): (A2,b2)o(A1,b1) = (A1*A2, A2*b1+b2).
// carry[k] = inclusive[k-1].b, carry[0] = 0.
// ---------------------------------------------------------------------------
__global__ __launch_bounds__(NCHUNK) void k2_carries(float* __restrict__ ws) {
  const float2* powA = (const float2*)(ws + OFF_POWA);
  const float2* cend = (const float2*)(ws + OFF_CEND);
  float2* carry      = (float2*)(ws + OFF_CARRY);

  __shared__ float sAr[NCHUNK], sAi[NCHUNK], sbr[NCHUNK], sbi[NCHUNK];

  const int p = blockIdx.x;              // gridDim == P
  const int t = threadIdx.x;             // blockDim == NCHUNK
  const float2 Alc = powA[(LC - 1) * P + p];  // Abar^LC
  const float2 e   = cend[t * P + p];
  float Ar = Alc.x, Ai = Alc.y, br = e.x, bi = e.y;
  sAr[t] = Ar; sAi[t] = Ai; sbr[t] = br; sbi[t] = bi;
  __syncthreads();

#pragma unroll
  for (int d = 1; d < NCHUNK; d <<= 1) {
    float pAr = 0.f, pAi = 0.f, pbr = 0.f, pbi = 0.f;
    if (t >= d) { pAr = sAr[t - d]; pAi = sAi[t - d];
                  pbr = sbr[t - d]; pbi = sbi[t - d]; }
    __syncthreads();
    if (t >= d) {
      const float nAr = Ar * pAr - Ai * pAi;      // A_t * A_{t-d}
      const float nAi = Ar * pAi + Ai * pAr;
      const float nbr = Ar * pbr - Ai * pbi + br; // A_t * b_{t-d} + b_t
      const float nbi = Ar * pbi + Ai * pbr + bi;
      Ar = nAr; Ai = nAi; br = nbr; bi = nbi;
      sAr[t] = Ar; sAi[t] = Ai; sbr[t] = br; sbi[t] = bi;
    }
    __syncthreads();
  }

  const float2 c = (t == 0) ? make_float2(0.f, 0.f)
                            : make_float2(sbr[t - 1], sbi[t - 1]);
  carry[t * P + p] = c;
}

// ---------------------------------------------------------------------------
// K3: splice carries into local x, then  y = x @ (2*C^T) + D*u  via WMMA.
//     x_global[l] = x_local[l] + Abar^(l_local+1) * carry_chunk.
// ---------------------------------------------------------------------------
__global__ __launch_bounds__(256) void k3_output(const float* __restrict__ u,
                                                 const float* __restrict__ D,
                                                 const float* __restrict__ ws,
                                                 float* __restrict__ out) {
  const float* CreT   = ws + OFF_CRET;
  const float* CimTn  = ws + OFF_CIMTN;
  const float2* powA  = (const float2*)(ws + OFF_POWA);
  const float2* carry = (const float2*)(ws + OFF_CARRY);
  const float2* xloc  = (const float2*)(ws + OFF_XLOC);

  __shared__ float  lds_xr[16 * (P + 2)];
  __shared__ float  lds_xi[16 * (P + 2)];
  __shared__ float2 lds_carry[P];

  const int chunk = blockIdx.x;
  const int tid   = threadIdx.x;
  const int wave  = tid >> 5;
  const int lane  = tid & 31;
  const int khalf = (lane >> 4) * 2;

  if (tid < P) lds_carry[tid] = carry[chunk * P + tid];

  for (int mt = 0; mt < LC / 16; ++mt) {
    __syncthreads();
    // load x tile, apply carry correction, stage in A-operand-friendly layout
    for (int i = tid; i < 16 * P; i += 256) {
      const int r  = i / P;
      const int pp = i % P;
      const int ll = mt * 16 + r;                       // l within chunk
      const float2 xl = xloc[(size_t)(chunk * LC + ll) * P + pp];
      const float2 pw = powA[ll * P + pp];              // Abar^(ll+1)
      const float2 cc = lds_carry[pp];
      lds_xr[r * (P + 2) + pp] = xl.x + (pw.x * cc.x - pw.y * cc.y);
      lds_xi[r * (P + 2) + pp] = xl.y + (pw.x * cc.y + pw.y * cc.x);
    }
    __syncthreads();

    // each wave produces 2 N-tiles of 16 output channels
    const float* xrrow = &lds_xr[(lane & 15) * (P + 2)];
    const float* xirow = &lds_xi[(lane & 15) * (P + 2)];
#pragma unroll
    for (int t = 0; t < 2; ++t) {
      const int n0 = (wave * 2 + t) * 16;
      const int h  = n0 + (lane & 15);
      v8f acc = {0.f, 0.f, 0.f, 0.f, 0.f, 0.f, 0.f, 0.f};
#pragma unroll 4
      for (int k0 = 0; k0 < P; k0 += 4) {
        v2f ar, ai, br_, bi_;
        ar.x = xrrow[k0 + khalf];  ar.y = xrrow[k0 + khalf + 1];
        ai.x = xirow[k0 + khalf];  ai.y = xirow[k0 + khalf + 1];
        const int rk = (k0 + khalf) * H + h;
        br_.x = CreT[rk];   br_.y = CreT[rk + H];    //  2*C_re^T
        bi_.x = CimTn[rk];  bi_.y = CimTn[rk + H];   // -2*C_im^T
        acc = wmma_f32(ar, br_, acc);   // += 2*xr@Cre^T
        acc = wmma_f32(ai, bi_, acc);   // -= 2*xi@Cim^T
      }
      // epilogue: + D*u, scatter per C/D VGPR layout
      const int rowoff = (lane >> 4) * 8;
      const float dh = D[h];
#pragma unroll
      for (int r = 0; r < 8; ++r) {
        const size_t gl = (size_t)chunk * LC + mt * 16 + rowoff + r;
        out[gl * H + h] = acc[r] + dh * u[gl * H + h];
      }
    }
  }
}

// ---------------------------------------------------------------------------
extern "C" void kernel_launch(void* const* d_in, const int* in_sizes, int n_in,
                              void* d_out, int out_size, void* d_ws,
                              size_t ws_size, hipStream_t stream) {
  const float* Lre   = (const float*)d_in[0];
  const float* Lim   = (const float*)d_in[1];
  const float* B     = (const float*)d_in[2];
  const float* C     = (const float*)d_in[3];
  const float* D     = (const float*)d_in[4];
  const float* lstep = (const float*)d_in[5];
  const float* u     = (const float*)d_in[6];
  float* out = (float*)d_out;
  float* ws  = (float*)d_ws;

  k0_setup<<<64, 256, 0, stream>>>(Lre, Lim, B, C, lstep, ws);
  k1_chunk<<<NCHUNK, 256, 0, stream>>>(u, ws);
  k2_carries<<<P, NCHUNK, 0, stream>>>(ws);
  k3_output<<<NCHUNK, 256, 0, stream>>>(u, D, ws, out);
}